// CenterLoss_17875653886475
// MI455X (gfx1250) — compile-verified
//
#include <hip/hip_runtime.h>

#define FEAT_DIM    512
#define NUM_CLASSES 50000
#define BATCH       8192
#define ALPHA       0.1f

typedef float v2f __attribute__((ext_vector_type(2)));
typedef float v8f __attribute__((ext_vector_type(8)));

// ---------------------------------------------------------------- zero counts
__global__ void k_zero_ws(float* __restrict__ ws, int n) {
    int i = blockIdx.x * blockDim.x + threadIdx.x;
    if (i < n) ws[i] = 0.0f;
}

// ---------------------------------------------------------------- class counts
__global__ void k_counts(const int* __restrict__ labels, float* __restrict__ counts) {
    int i = blockIdx.x * blockDim.x + threadIdx.x;
    if (i < BATCH) atomicAdd(&counts[labels[i]], 1.0f);  // exact: sums of 1.0
}

// ---------------------------------------------------------------- center loss via WMMA row-sum
// One wave owns 16 batch rows. A(16x4) = squared diffs, B(4x16) = ones, so the
// matrix pipe accumulates per-row sums of (f-c)^2 into the f32 16x16 accumulator.
// A layout (32-bit 16x4): lanes 0-15 -> (M=lane, K=k+0,k+1); lanes 16-31 -> (M=lane-16, K=k+2,k+3).
__global__ __launch_bounds__(256) void k_loss_wmma(const float* __restrict__ feat,
                                                   const int*   __restrict__ labels,
                                                   const float* __restrict__ centers,
                                                   float*       __restrict__ partials) {
    const int lane = threadIdx.x & 31;
    const int wave = (blockIdx.x * blockDim.x + threadIdx.x) >> 5;   // 0..511
    const int row  = (wave << 4) + (lane & 15);
    const int half = lane >> 4;
    const int lab  = labels[row];

    const float* frow = feat    + (size_t)row * FEAT_DIM;
    const float* crow = centers + (size_t)lab * FEAT_DIM;

    v8f acc = {};
    v2f ones; ones.x = 1.0f; ones.y = 1.0f;

    for (int k = 0; k < FEAT_DIM; k += 4) {
        const int col = k + 2 * half;                 // 8-byte aligned
        v2f fv = *(const v2f*)(frow + col);
        v2f cv = *(const v2f*)(crow + col);
        v2f a;
        float d0 = fv.x - cv.x, d1 = fv.y - cv.y;
        a.x = d0 * d0; a.y = d1 * d1;
        acc = __builtin_amdgcn_wmma_f32_16x16x4_f32(
            /*neg_a=*/false, a, /*neg_b=*/false, ones,
            /*c_mod=*/(short)0, acc, /*reuse_a=*/false, /*reuse_b=*/false);
    }

    // D[i][j] identical over j: lanes 0-15 hold rows M=0..7 in acc[0..7],
    // lanes 16-31 hold rows M=8..15. Combine halves, lane 0 emits tile sum.
    float local = acc[0] + acc[1] + acc[2] + acc[3] + acc[4] + acc[5] + acc[6] + acc[7];
    float tot = local + __shfl_xor(local, 16, 32);
    if (lane == 0) partials[wave] = tot;
}

// ---------------------------------------------------------------- streaming centers pass
// out = centers * (present ? (1-ALPHA) : 1).  The 100MB read + 100MB write pass.
__global__ void k_update_init(const float* __restrict__ centers,
                              const float* __restrict__ counts,
                              float*       __restrict__ out) {
    size_t t    = (size_t)blockIdx.x * blockDim.x + threadIdx.x;
    size_t base = t << 2;                                   // 4 elems / thread
    if (base >= (size_t)NUM_CLASSES * FEAT_DIM) return;
    int   c     = (int)(base >> 9);                         // FEAT_DIM = 512
    float cnt   = counts[c];
    float scale = (cnt > 0.0f) ? (1.0f - ALPHA) : 1.0f;
    float4 cv   = *(const float4*)(centers + base);         // aligned b128 load
    out[base + 0] = cv.x * scale;                           // out is 4B-aligned only
    out[base + 1] = cv.y * scale;
    out[base + 2] = cv.z * scale;
    out[base + 3] = cv.w * scale;
}

// ---------------------------------------------------------------- EMA scatter
// out[lab, d] += ALPHA * f[i, d] / count[lab]   (only present classes touched)
__global__ void k_scatter(const float* __restrict__ feat,
                          const int*   __restrict__ labels,
                          const float* __restrict__ counts,
                          float*       __restrict__ out) {
    size_t t = (size_t)blockIdx.x * blockDim.x + threadIdx.x;
    if (t >= (size_t)BATCH * (FEAT_DIM / 4)) return;
    int i  = (int)(t >> 7);                 // FEAT_DIM/4 = 128
    int d0 = (int)(t & 127) << 2;
    int lab = labels[i];
    float w = ALPHA / counts[lab];          // count >= 1 here
    float4 fv = *(const float4*)(feat + (size_t)i * FEAT_DIM + d0);  // aligned
    float* o = out + (size_t)lab * FEAT_DIM + d0;
    atomicAdd(o + 0, fv.x * w);
    atomicAdd(o + 1, fv.y * w);
    atomicAdd(o + 2, fv.z * w);
    atomicAdd(o + 3, fv.w * w);
}

// ---------------------------------------------------------------- deterministic loss finalize
__global__ void k_loss_final(const float* __restrict__ partials, float* __restrict__ out_loss) {
    __shared__ float sm[256];
    int t = threadIdx.x;
    sm[t] = partials[t] + partials[t + 256];   // 512 partials
    __syncthreads();
    for (int s = 128; s > 0; s >>= 1) {
        if (t < s) sm[t] += sm[t + s];
        __syncthreads();
    }
    if (t == 0) out_loss[0] = sm[0] / (float)BATCH;
}

extern "C" void kernel_launch(void* const* d_in, const int* in_sizes, int n_in,
                              void* d_out, int out_size, void* d_ws, size_t ws_size,
                              hipStream_t stream) {
    const float* feat    = (const float*)d_in[0];   // [8192, 512]
    const int*   labels  = (const int*)  d_in[1];   // [8192]
    const float* centers = (const float*)d_in[2];   // [50000, 512]

    float* out      = (float*)d_out;        // out[0] = loss
    float* out_nc   = out + 1;              // out[1..] = new_centers (4B aligned)

    float* counts   = (float*)d_ws;                 // [NUM_CLASSES]
    float* partials = counts + NUM_CLASSES;         // [512]

    (void)in_sizes; (void)n_in; (void)out_size; (void)ws_size;

    // 1) zero class counts
    k_zero_ws<<<(NUM_CLASSES + 255) / 256, 256, 0, stream>>>(counts, NUM_CLASSES);

    // 2) per-class counts
    k_counts<<<(BATCH + 255) / 256, 256, 0, stream>>>(labels, counts);

    // 3) center loss via WMMA (512 waves, 16 rows each -> 64 blocks x 256)
    k_loss_wmma<<<64, 256, 0, stream>>>(feat, labels, centers, partials);

    // 4) streaming pass: out = centers * (present ? 1-ALPHA : 1)
    {
        size_t n4 = (size_t)NUM_CLASSES * FEAT_DIM / 4;   // 6,400,000
        k_update_init<<<(unsigned)((n4 + 255) / 256), 256, 0, stream>>>(centers, counts, out_nc);
    }

    // 5) scatter ALPHA * f / count into present rows
    {
        size_t n4 = (size_t)BATCH * FEAT_DIM / 4;         // 1,048,576
        k_scatter<<<(unsigned)((n4 + 255) / 256), 256, 0, stream>>>(feat, labels, counts, out_nc);
    }

    // 6) deterministic loss reduction
    k_loss_final<<<1, 256, 0, stream>>>(partials, out);
}